// LambdaLayer2D_43516608643265
// MI455X (gfx1250) — compile-verified
//
#include <hip/hip_runtime.h>
#include <cmath>

typedef __attribute__((ext_vector_type(2))) float v2f;
typedef __attribute__((ext_vector_type(8))) float v8f;

#define B_     16
#define C_     256
#define HW_    64
#define N_     4096
#define KD_    16
#define HEADS_ 4
#define VD_    64
#define OCH_   144   // 64 q-rows, 16 k-rows, 64 v-rows

__device__ __forceinline__ v8f wmma_f32_16x16x4(v2f a, v2f b, v8f c) {
    // D = A(16x4,f32) * B(4x16,f32) + C(16x16,f32), wave32
    return __builtin_amdgcn_wmma_f32_16x16x4_f32(false, a, false, b, (short)0, c, false, false);
}

__device__ __forceinline__ const float* wrow(const float* Wq, const float* Wk,
                                             const float* Wv, int r) {
    if (r < 64) return Wq + (size_t)r * C_;
    if (r < 80) return Wk + (size_t)(r - 64) * C_;
    return Wv + (size_t)(r - 80) * C_;
}

// ---------------------------------------------------------------------------
// Kernel 1: P[b][o][n] = sum_c W_all[o][c] * x[b][c][n]   (WMMA f32 16x16x4)
// grid (N/128, B), block 256 (8 waves). Each wave owns one 16-wide n-tile and
// computes ALL 9 o-tiles (144 rows) so x is streamed exactly once. Per k-step:
// batch-load 1 B fragment + 9 A fragments (load clause), then fire 9
// back-to-back WMMAs so the loads drain under graduated s_wait_loadcnt.
// ---------------------------------------------------------------------------
__global__ void proj_gemm(const float* __restrict__ x, const float* __restrict__ Wq,
                          const float* __restrict__ Wk, const float* __restrict__ Wv,
                          float* __restrict__ P) {
    const int lane = threadIdx.x & 31;
    const int wave = threadIdx.x >> 5;
    const int b  = blockIdx.y;
    const int n0 = blockIdx.x * 128 + wave * 16;
    const int mrow = lane & 15;
    const int koff = (lane >> 4) << 1;            // K pair: 0/1 (lanes 0-15), 2/3 (16-31)

    const float* Arows[9];
    #pragma unroll
    for (int o = 0; o < 9; ++o)
        Arows[o] = wrow(Wq, Wk, Wv, o * 16 + mrow) + koff;
    const float* Bcol = x + (size_t)b * C_ * N_ + n0 + mrow;   // col fixed, row stride N_

    v8f acc[9] = {};
    for (int k = 0; k < C_; k += 4) {
        v2f bb;
        bb.x = Bcol[(size_t)(k + koff) * N_];
        bb.y = Bcol[(size_t)(k + koff + 1) * N_];
        v2f a[9];
        #pragma unroll
        for (int o = 0; o < 9; ++o) {             // one big load clause
            a[o].x = Arows[o][k];
            a[o].y = Arows[o][k + 1];
        }
        #pragma unroll
        for (int o = 0; o < 9; ++o)               // back-to-back WMMA burst
            acc[o] = wmma_f32_16x16x4(a[o], bb, acc[o]);
    }

    const int rshift = (lane >> 4) << 3;          // D: M = j (+8 for upper lanes)
    #pragma unroll
    for (int o = 0; o < 9; ++o) {
        float* Pout = P + ((size_t)b * OCH_ + o * 16) * N_ + n0 + mrow;
        #pragma unroll
        for (int j = 0; j < 8; ++j)
            Pout[(size_t)(j + rshift) * N_] = acc[o][j];
    }
}

// ---------------------------------------------------------------------------
// Kernel 2: BatchNorm batch statistics -> per-channel scale/shift
// ---------------------------------------------------------------------------
__global__ void bn_stats(const float* __restrict__ P,
                         const float* __restrict__ gq, const float* __restrict__ bq,
                         const float* __restrict__ gv, const float* __restrict__ bv,
                         float* __restrict__ sc_q, float* __restrict__ sh_q,
                         float* __restrict__ sc_v, float* __restrict__ sh_v) {
    __shared__ float s1[256], s2[256];
    const int ch  = blockIdx.x;                     // 0..127
    const int row = (ch < 64) ? ch : (80 + (ch - 64));
    const int t   = threadIdx.x;
    float sum = 0.f, ssq = 0.f;
    for (int i = t; i < B_ * N_; i += 256) {
        int b = i >> 12, n = i & (N_ - 1);
        float v = P[((size_t)b * OCH_ + row) * N_ + n];
        sum += v; ssq += v * v;
    }
    s1[t] = sum; s2[t] = ssq;
    __syncthreads();
    for (int off = 128; off > 0; off >>= 1) {
        if (t < off) { s1[t] += s1[t + off]; s2[t] += s2[t + off]; }
        __syncthreads();
    }
    if (t == 0) {
        const float inv_n = 1.f / (float)(B_ * N_);
        float mean = s1[0] * inv_n;
        float var  = s2[0] * inv_n - mean * mean;   // biased, matches reference
        float inv  = rsqrtf(var + 1e-5f);
        if (ch < 64) {
            float sc = gq[ch] * inv;
            sc_q[ch] = sc; sh_q[ch] = bq[ch] - mean * sc;
        } else {
            int c2 = ch - 64;
            float sc = gv[c2] * inv;
            sc_v[c2] = sc; sh_v[c2] = bv[c2] - mean * sc;
        }
    }
}

// ---------------------------------------------------------------------------
// Kernel 3: softmax row stats (max, 1/sum) per (b,k) over n
// ---------------------------------------------------------------------------
__global__ void softmax_stats(const float* __restrict__ P,
                              float* __restrict__ rmax, float* __restrict__ rrsum) {
    __shared__ float sh[256];
    const int bk = blockIdx.x;
    const int b = bk >> 4, kk = bk & 15;
    const float* row = P + ((size_t)b * OCH_ + 64 + kk) * N_;
    const int t = threadIdx.x;
    float m = -1e30f;
    for (int n = t; n < N_; n += 256) m = fmaxf(m, row[n]);
    sh[t] = m; __syncthreads();
    for (int off = 128; off > 0; off >>= 1) {
        if (t < off) sh[t] = fmaxf(sh[t], sh[t + off]);
        __syncthreads();
    }
    float mf = sh[0];
    __syncthreads();
    float s = 0.f;
    for (int n = t; n < N_; n += 256) s += expf(row[n] - mf);
    sh[t] = s; __syncthreads();
    for (int off = 128; off > 0; off >>= 1) {
        if (t < off) sh[t] += sh[t + off];
        __syncthreads();
    }
    if (t == 0) { rmax[bk] = mf; rrsum[bk] = 1.f / sh[0]; }
}

// ---------------------------------------------------------------------------
// Kernel 4: lambda_c[b][k][v] = sum_m softmax(k)[b,k,m] * v_bn[b,v,m]
// WMMA f32 16x16x4; 8 waves stripe m, LDS cross-wave reduction.
// grid (VD/16, B), block 256
// ---------------------------------------------------------------------------
__global__ void lambda_c_gemm(const float* __restrict__ P,
                              const float* __restrict__ rmax, const float* __restrict__ rrsum,
                              const float* __restrict__ sc_v, const float* __restrict__ sh_v,
                              float* __restrict__ lamc) {
    __shared__ float red[8 * 256];
    const int lane = threadIdx.x & 31;
    const int wave = threadIdx.x >> 5;
    const int b  = blockIdx.y;
    const int v0 = blockIdx.x * 16;
    const int kk   = lane & 15;
    const int koff = (lane >> 4) << 1;
    const float rm = rmax[b * 16 + kk];
    const float rs = rrsum[b * 16 + kk];
    const int vcol = v0 + (lane & 15);
    const float scv = sc_v[vcol], shv = sh_v[vcol];
    const float* Ak = P + ((size_t)b * OCH_ + 64 + kk) * N_;
    const float* Bv = P + ((size_t)b * OCH_ + 80 + vcol) * N_;

    v8f acc = {};
    for (int i = 0; i < N_ / 32; ++i) {
        const int m = i * 32 + wave * 4 + koff;
        v2f a, bb;
        a.x  = expf(Ak[m] - rm) * rs;
        a.y  = expf(Ak[m + 1] - rm) * rs;
        bb.x = Bv[m] * scv + shv;
        bb.y = Bv[m + 1] * scv + shv;
        acc = wmma_f32_16x16x4(a, bb, acc);
    }

    // D layout: k = j + 8*(lane>=16), v_local = lane&15
    const int vloc = lane & 15;
    const int madd = (lane >> 4) << 3;
    #pragma unroll
    for (int j = 0; j < 8; ++j)
        red[wave * 256 + (j + madd) * 16 + vloc] = acc[j];
    __syncthreads();
    const int t = threadIdx.x;
    float s = 0.f;
    #pragma unroll
    for (int w = 0; w < 8; ++w) s += red[w * 256 + t];
    lamc[((size_t)b * 16 + (t >> 4)) * 64 + v0 + (t & 15)] = s;
}

// ---------------------------------------------------------------------------
// Kernel 5: fused output  Y[b,h,v,n] = sum_k q_bn[b,h,k,n] * (lamc[b,k,v]
//             + conv7x7(v_bn)[k,v,n] + pos_b[k])
// refactored: s[t] = sum_k q*pos_w[k,t];  y += sum_t s[t]*v_bn(shifted).
// Loop order h-outer / v-chunk-inner: s[49] computed once per head (min FMA);
// the 16-channel v halo tile is refilled per (h,chunk) from L2-resident data,
// with prefetch of the next chunk. grid (W/16, H/16, B), block 256.
// ---------------------------------------------------------------------------
__global__ void final_fused(const float* __restrict__ P,
                            const float* __restrict__ sc_q, const float* __restrict__ sh_q,
                            const float* __restrict__ sc_v, const float* __restrict__ sh_v,
                            const float* __restrict__ lamc,
                            const float* __restrict__ pos_w, const float* __restrict__ pos_b,
                            float* __restrict__ out) {
    __shared__ float vt[16 * 484];     // 16 v-channels x (22x22) halo tile
    __shared__ float lamS[16 * 64];
    __shared__ float pwS[49 * 16];     // [tap][k]
    __shared__ float pbS[16];
    __shared__ float sqS[64], shS[64], svS[64], svhS[64];

    const int t  = threadIdx.x;
    const int b  = blockIdx.z;
    const int R0 = blockIdx.y * 16;
    const int C0 = blockIdx.x * 16;
    const int r = t >> 4, c = t & 15;
    const int n = (R0 + r) * HW_ + (C0 + c);

    for (int i = t; i < 1024; i += 256) lamS[i] = lamc[(size_t)b * 1024 + i];
    for (int i = t; i < 784; i += 256) {
        int k = i / 49, tp = i % 49;
        pwS[tp * 16 + k] = pos_w[i];               // pos_w flat (K,1,7,7)
    }
    if (t < 16) pbS[t] = pos_b[t];
    if (t < 64) { sqS[t] = sc_q[t]; shS[t] = sh_q[t]; svS[t] = sc_v[t]; svhS[t] = sh_v[t]; }
    __syncthreads();

    // q (BN applied) for this position, all 64 channels, kept in registers
    float qall[64];
    #pragma unroll
    for (int ch = 0; ch < 64; ++ch)
        qall[ch] = P[((size_t)b * OCH_ + ch) * N_ + n] * sqS[ch] + shS[ch];

    #pragma unroll
    for (int h = 0; h < 4; ++h) {
        // s[t] = sum_k q[h,k] * pos_w[k,t] : computed once per head
        float s[49];
        #pragma unroll
        for (int tp = 0; tp < 49; ++tp) {
            float a = 0.f;
            #pragma unroll
            for (int k = 0; k < 16; ++k) a += qall[h * 16 + k] * pwS[tp * 16 + k];
            s[tp] = a;
        }
        float qb = 0.f;
        #pragma unroll
        for (int k = 0; k < 16; ++k) qb += qall[h * 16 + k] * pbS[k];

        for (int vc = 0; vc < 4; ++vc) {
            __syncthreads();
            for (int i = t; i < 16 * 484; i += 256) {
                const int vl = i / 484, p = i % 484;
                const int hy = R0 - 3 + p / 22, hx = C0 - 3 + p % 22;
                const int vg = vc * 16 + vl;
                float val = 0.f;                   // zero ('SAME') padding
                const bool inb = (hy >= 0 && hy < HW_ && hx >= 0 && hx < HW_);
                if (inb) {
                    const float* src = &P[((size_t)b * OCH_ + 80 + vg) * N_ + hy * HW_ + hx];
                    val = *src * svS[vg] + svhS[vg];
                    // prefetch next v-chunk's line (L2 -> near cache)
                    if (vc < 3 && (i & 7) == 0)
                        __builtin_prefetch(src + 16 * (size_t)N_, 0, 1);
                }
                vt[i] = val;
            }
            __syncthreads();

            for (int vl = 0; vl < 16; ++vl) {
                const int vg = vc * 16 + vl;
                float acc = qb;
                #pragma unroll
                for (int k = 0; k < 16; ++k) acc += qall[h * 16 + k] * lamS[k * 64 + vg];
                const float* vbase = &vt[vl * 484 + r * 22 + c];
                #pragma unroll
                for (int tp = 0; tp < 49; ++tp)
                    acc += s[tp] * vbase[(tp / 7) * 22 + (tp % 7)];
                out[((size_t)b * 256 + h * 64 + vg) * (size_t)N_ + n] = acc;
            }
        }
    }
}

// ---------------------------------------------------------------------------
extern "C" void kernel_launch(void* const* d_in, const int* in_sizes, int n_in,
                              void* d_out, int out_size, void* d_ws, size_t ws_size,
                              hipStream_t stream) {
    const float* x    = (const float*)d_in[0];
    const float* Wq   = (const float*)d_in[1];
    const float* Wk   = (const float*)d_in[2];
    const float* Wv   = (const float*)d_in[3];
    const float* gq   = (const float*)d_in[4];
    const float* bq   = (const float*)d_in[5];
    const float* gv   = (const float*)d_in[6];
    const float* bv   = (const float*)d_in[7];
    const float* posw = (const float*)d_in[8];
    const float* posb = (const float*)d_in[9];
    float* out = (float*)d_out;

    float* ws = (float*)d_ws;
    float* P     = ws;                                  // B*144*N floats (~37.7 MB)
    size_t Psz   = (size_t)B_ * OCH_ * N_;
    float* sc_q  = P + Psz;
    float* sh_q  = sc_q + 64;
    float* sc_v  = sh_q + 64;
    float* sh_v  = sc_v + 64;
    float* rmax  = sh_v + 64;
    float* rrsum = rmax + 256;
    float* lamc  = rrsum + 256;                         // B*16*64 floats

    proj_gemm<<<dim3(N_ / 128, B_), 256, 0, stream>>>(x, Wq, Wk, Wv, P);
    bn_stats<<<128, 256, 0, stream>>>(P, gq, bq, gv, bv, sc_q, sh_q, sc_v, sh_v);
    softmax_stats<<<256, 256, 0, stream>>>(P, rmax, rrsum);
    lambda_c_gemm<<<dim3(VD_ / 16, B_), 256, 0, stream>>>(P, rmax, rrsum, sc_v, sh_v, lamc);
    final_fused<<<dim3(HW_ / 16, HW_ / 16, B_), 256, 0, stream>>>(
        P, sc_q, sh_q, sc_v, sh_v, lamc, posw, posb, out);
}